// GCN_86260123174490
// MI455X (gfx1250) — compile-verified
//
#include <hip/hip_runtime.h>
#include <hip/hip_bf16.h>
#include <math.h>
#include <limits.h>

// ---------------------------------------------------------------------------
// DGCNN pipeline for MI455X (gfx1250, wave32).
// Big GEMMs use V_WMMA_F32_16X16X4_F32 (exact fp32 on the matrix pipe).
// Edge scatter (the HBM-bound part, ~1.6 GB of traffic) uses coalesced
// global f32 atomics.
// ---------------------------------------------------------------------------

#define F_DIM   512
#define H_DIM   512
#define KSEL    30
#define NCH1    16
#define NCH2    32
#define CONV2_LEN 11
#define FLAT_DIM  352
#define NCLS    10

typedef __attribute__((ext_vector_type(2))) float v2f;
typedef __attribute__((ext_vector_type(8))) float v8f;

// ---------------------------------------------------------------------------
// C(M x 512) = A(M x 512) * B(512 x 512), fp32, via V_WMMA_F32_16X16X4_F32.
// Block: 256 threads = 8 waves. Block tile: 16 (M) x 128 (N); each wave owns
// one 16x16 C tile. K staged through LDS in 64-wide chunks.
// ---------------------------------------------------------------------------
__global__ __launch_bounds__(256) void gemm512_wmma(const float* __restrict__ A,
                                                    const float* __restrict__ B,
                                                    float* __restrict__ C,
                                                    int M) {
    const int NDIM = 512, KDIM = 512, TK = 64;
    __shared__ float As[16][TK + 1];    // +1 pad: kill column bank conflicts
    __shared__ float Bs[TK][128 + 4];   // +4 pad

    const int tid  = threadIdx.x;
    const int wave = tid >> 5;
    const int lane = tid & 31;
    const int m0      = blockIdx.y * 16;
    const int n_block = blockIdx.x * 128;

    const int mrow  = lane & 15;         // M (for A) / N (for B) index in tile
    const int khalf = (lane >> 4) * 2;   // lanes 16-31 carry K+2, K+3

    v8f acc = {};

    for (int k0 = 0; k0 < KDIM; k0 += TK) {
        // Stage A tile: 16 x 64
        for (int i = tid; i < 16 * TK; i += 256) {
            int r = i / TK, c = i % TK;
            int row = m0 + r; if (row >= M) row = M - 1;   // clamp (unused rows)
            As[r][c] = A[(size_t)row * KDIM + k0 + c];
        }
        // Stage B tile: 64 x 128
        for (int i = tid; i < TK * 128; i += 256) {
            int r = i >> 7, c = i & 127;
            Bs[r][c] = B[(size_t)(k0 + r) * NDIM + n_block + c];
        }
        __syncthreads();

        #pragma unroll
        for (int kk = 0; kk < TK; kk += 4) {
            v2f a, b;
            a.x = As[mrow][kk + khalf];
            a.y = As[mrow][kk + khalf + 1];
            b.x = Bs[kk + khalf][wave * 16 + mrow];
            b.y = Bs[kk + khalf + 1][wave * 16 + mrow];
            acc = __builtin_amdgcn_wmma_f32_16x16x4_f32(
                      false, a, false, b, (short)0, acc, false, false);
        }
        __syncthreads();
    }

    // Store: VGPR r -> M = r (lanes 0-15) or 8+r (lanes 16-31), N = lane%16
    const int nn    = lane & 15;
    const int mbase = (lane >> 4) * 8;
    #pragma unroll
    for (int r = 0; r < 8; ++r) {
        int row = m0 + mbase + r;
        if (row < M)
            C[(size_t)row * NDIM + n_block + wave * 16 + nn] = acc[r];
    }
}

// ---------------------------------------------------------------------------
// Degree / normalization
// ---------------------------------------------------------------------------
__global__ void fill_f32(float* p, float v, int n) {
    int i = blockIdx.x * blockDim.x + threadIdx.x;
    if (i < n) p[i] = v;
}
__global__ void deg_edges(float* deg, const int* __restrict__ dst, int n_edges) {
    int e = blockIdx.x * blockDim.x + threadIdx.x;
    if (e < n_edges) atomicAdd(&deg[dst[e]], 1.0f);
}
__global__ void deg_to_dinv(float* deg, int n) {
    int i = blockIdx.x * blockDim.x + threadIdx.x;
    if (i < n) { float d = deg[i]; deg[i] = (d > 0.0f) ? (1.0f / sqrtf(d)) : 0.0f; }
}

// acc[i,c] = bias[c] + h[i,c] * dinv[i]^2   (self-loop term + bias)
__global__ void init_acc512(float* __restrict__ acc, const float* __restrict__ h,
                            const float* __restrict__ dinv, const float* __restrict__ bias,
                            int n_nodes) {
    size_t t = blockIdx.x * (size_t)blockDim.x + threadIdx.x;
    if (t >= (size_t)n_nodes * 512) return;
    int i = (int)(t >> 9), c = (int)(t & 511);
    float di = dinv[i];
    acc[t] = bias[c] + h[t] * di * di;
}

// acc[dst,c] += h[src,c] * dinv[src]*dinv[dst]    (coalesced over c)
__global__ void scatter512(const float* __restrict__ h, const int* __restrict__ src,
                           const int* __restrict__ dst, const float* __restrict__ dinv,
                           float* __restrict__ acc, int n_edges) {
    size_t t = blockIdx.x * (size_t)blockDim.x + threadIdx.x;
    if (t >= (size_t)n_edges * 512) return;
    int e = (int)(t >> 9), c = (int)(t & 511);
    int s = src[e], d = dst[e];
    float nrm = dinv[s] * dinv[d];
    atomicAdd(&acc[(size_t)d * 512 + c], h[(size_t)s * 512 + c] * nrm);
}

// out = elu(acc [+ resid])
__global__ void elu_kernel(float* __restrict__ out, const float* __restrict__ acc,
                           const float* __restrict__ resid, size_t n) {
    size_t i = blockIdx.x * (size_t)blockDim.x + threadIdx.x;
    if (i >= n) return;
    float v = acc[i] + (resid ? resid[i] : 0.0f);
    out[i] = (v > 0.0f) ? v : expm1f(v);
}

// h3[i] = dot(x2[i,:], W3)  — one wave per node, coalesced
__global__ void matvec_w3(const float* __restrict__ x2, const float* __restrict__ W3,
                          float* __restrict__ h3, int n_nodes) {
    int gt = blockIdx.x * blockDim.x + threadIdx.x;
    int wid = gt >> 5, lane = gt & 31;
    if (wid >= n_nodes) return;
    const float* row = x2 + (size_t)wid * H_DIM;
    float s = 0.0f;
    for (int j = lane; j < H_DIM; j += 32) s += row[j] * W3[j];
    for (int off = 16; off; off >>= 1) s += __shfl_xor(s, off, 32);
    if (lane == 0) h3[wid] = s;
}

__global__ void out3_init(float* __restrict__ out3, const float* __restrict__ h3,
                          const float* __restrict__ dinv, const float* __restrict__ b3,
                          int n_nodes) {
    int i = blockIdx.x * blockDim.x + threadIdx.x;
    if (i >= n_nodes) return;
    float di = dinv[i];
    out3[i] = b3[0] + h3[i] * di * di;
}
__global__ void scatter_scalar(const float* __restrict__ h3, const int* __restrict__ src,
                               const int* __restrict__ dst, const float* __restrict__ dinv,
                               float* __restrict__ out3, int n_edges) {
    int e = blockIdx.x * blockDim.x + threadIdx.x;
    if (e >= n_edges) return;
    int s = src[e], d = dst[e];
    atomicAdd(&out3[d], h3[s] * dinv[s] * dinv[d]);
}

// ---------------------------------------------------------------------------
// Sort-pool bookkeeping (batch is pre-sorted ascending -> contiguous ranges)
// ---------------------------------------------------------------------------
__global__ void zero_i32(int* p, int n) {
    int i = blockIdx.x * blockDim.x + threadIdx.x;
    if (i < n) p[i] = 0;
}
__global__ void hist_batch(const int* __restrict__ batch, int* __restrict__ counts, int n_nodes) {
    int i = blockIdx.x * blockDim.x + threadIdx.x;
    if (i < n_nodes) atomicAdd(&counts[batch[i]], 1);
}
__global__ void scan_starts(const int* __restrict__ counts, int* __restrict__ starts, int n_graphs) {
    if (threadIdx.x == 0 && blockIdx.x == 0) {
        int a = 0;
        for (int g = 0; g < n_graphs; ++g) { starts[g] = a; a += counts[g]; }
    }
}

// One wave per graph: iterative top-KSEL selection by (key desc, idx asc),
// which exactly reproduces a stable descending sort's first K entries.
__global__ void topk_kernel(const float* __restrict__ key, const int* __restrict__ starts,
                            const int* __restrict__ counts, int* __restrict__ sel) {
    int g = blockIdx.x;
    int lane = threadIdx.x;
    int s = starts[g], cnt = counts[g];
    float prevK = INFINITY;
    int   prevI = -1;
    for (int r = 0; r < KSEL; ++r) {
        float bk = -INFINITY; int bi = INT_MAX;
        for (int j = lane; j < cnt; j += 32) {
            int i = s + j;
            float kv = key[i];
            bool cand = (kv < prevK) || (kv == prevK && i > prevI);
            if (cand && (kv > bk || (kv == bk && i < bi))) { bk = kv; bi = i; }
        }
        for (int off = 16; off; off >>= 1) {
            float ok = __shfl_xor(bk, off, 32);
            int   oi = __shfl_xor(bi, off, 32);
            if (ok > bk || (ok == bk && oi < bi)) { bk = ok; bi = oi; }
        }
        int v = (r < cnt && bi != INT_MAX) ? bi : -1;
        if (lane == 0) sel[g * KSEL + r] = v;
        if (v >= 0) { prevK = bk; prevI = bi; }
    }
}

// ---------------------------------------------------------------------------
// conv1 (window == stride == 1025 -> one pooled row per position); gathers
// feature row [x1 | x2 | out3] directly from the selected node. z1: (g,c,t)
// ---------------------------------------------------------------------------
__global__ void conv1_relu(const float* __restrict__ x1, const float* __restrict__ x2,
                           const float* __restrict__ out3, const int* __restrict__ sel,
                           const float* __restrict__ cw1, const float* __restrict__ cb1,
                           float* __restrict__ z1, int n_graphs) {
    int idx = blockIdx.x * blockDim.x + threadIdx.x;
    int total = n_graphs * KSEL * NCH1;
    if (idx >= total) return;
    int c  = idx & (NCH1 - 1);
    int gt = idx >> 4;
    int t  = gt % KSEL;
    int g  = gt / KSEL;
    int node = sel[g * KSEL + t];
    float sum = cb1[c];
    if (node >= 0) {
        const float* w = cw1 + (size_t)c * (2 * H_DIM + 1);
        const float* a = x1 + (size_t)node * H_DIM;
        const float* b = x2 + (size_t)node * H_DIM;
        float acc = 0.0f;
        for (int j = 0; j < H_DIM; ++j) acc += w[j] * a[j];
        for (int j = 0; j < H_DIM; ++j) acc += w[H_DIM + j] * b[j];
        acc += w[2 * H_DIM] * out3[node];
        sum += acc;
    }
    z1[((size_t)g * NCH1 + c) * KSEL + t] = fmaxf(sum, 0.0f);
}

// fused maxpool(2) + conv2(k=5) + relu + flatten -> flat: (g, co*11 + t)
__global__ void conv2_pool_relu(const float* __restrict__ z1, const float* __restrict__ cw2,
                                const float* __restrict__ cb2, float* __restrict__ flat,
                                int n_graphs) {
    int idx = blockIdx.x * blockDim.x + threadIdx.x;
    int total = n_graphs * NCH2 * CONV2_LEN;
    if (idx >= total) return;
    int t  = idx % CONV2_LEN;
    int co = (idx / CONV2_LEN) % NCH2;
    int g  = idx / (CONV2_LEN * NCH2);
    float sum = cb2[co];
    for (int ci = 0; ci < NCH1; ++ci) {
        const float* zrow = z1 + ((size_t)g * NCH1 + ci) * KSEL;
        const float* w    = cw2 + ((size_t)co * NCH1 + ci) * 5;
        #pragma unroll
        for (int k = 0; k < 5; ++k) {
            int p = t + k;                       // pooled position 0..14
            float v = fmaxf(zrow[2 * p], zrow[2 * p + 1]);
            sum += w[k] * v;
        }
    }
    flat[(size_t)g * FLAT_DIM + co * CONV2_LEN + t] = fmaxf(sum, 0.0f);
}

__global__ void dense1_relu(const float* __restrict__ flat, const float* __restrict__ lw1,
                            const float* __restrict__ lb1, float* __restrict__ r1,
                            int n_graphs) {
    int idx = blockIdx.x * blockDim.x + threadIdx.x;
    if (idx >= n_graphs * H_DIM) return;
    int j = idx & (H_DIM - 1);
    int g = idx >> 9;
    const float* f = flat + (size_t)g * FLAT_DIM;
    float sum = lb1[j];
    for (int i = 0; i < FLAT_DIM; ++i) sum += f[i] * lw1[(size_t)i * H_DIM + j];
    r1[(size_t)g * H_DIM + j] = fmaxf(sum, 0.0f);
}

__global__ void dense2(const float* __restrict__ r1, const float* __restrict__ lw2,
                       const float* __restrict__ lb2, float* __restrict__ out,
                       int n_graphs) {
    int idx = blockIdx.x * blockDim.x + threadIdx.x;
    if (idx >= n_graphs * NCLS) return;
    int c = idx % NCLS;
    int g = idx / NCLS;
    const float* r = r1 + (size_t)g * H_DIM;
    float sum = lb2[c];
    for (int j = 0; j < H_DIM; ++j) sum += r[j] * lw2[(size_t)j * NCLS + c];
    out[(size_t)g * NCLS + c] = sum;
}

// ---------------------------------------------------------------------------
extern "C" void kernel_launch(void* const* d_in, const int* in_sizes, int n_in,
                              void* d_out, int out_size, void* d_ws, size_t ws_size,
                              hipStream_t stream) {
    (void)n_in; (void)ws_size;
    const float* x    = (const float*)d_in[0];
    const int*   eidx = (const int*)d_in[1];
    const int*   batch= (const int*)d_in[2];
    const float* W1 = (const float*)d_in[3];
    const float* b1 = (const float*)d_in[4];
    const float* W2 = (const float*)d_in[5];
    const float* b2 = (const float*)d_in[6];
    const float* W3 = (const float*)d_in[7];
    const float* b3 = (const float*)d_in[8];
    const float* cw1= (const float*)d_in[9];
    const float* cb1= (const float*)d_in[10];
    const float* cw2= (const float*)d_in[11];
    const float* cb2= (const float*)d_in[12];
    const float* lw1= (const float*)d_in[13];
    const float* lb1= (const float*)d_in[14];
    const float* lw2= (const float*)d_in[15];
    const float* lb2= (const float*)d_in[16];

    const int n_nodes  = in_sizes[0] / F_DIM;
    const int n_edges  = in_sizes[1] / 2;
    const int n_graphs = out_size / NCLS;
    const int* src = eidx;
    const int* dst = eidx + n_edges;

    // ---- workspace carve (256B aligned) ----
    char* p = (char*)d_ws;
    auto carve = [&](size_t bytes) { char* r = p; p += (bytes + 255) & ~(size_t)255; return (void*)r; };
    float* dinv = (float*)carve((size_t)n_nodes * 4);
    float* hbuf = (float*)carve((size_t)n_nodes * H_DIM * 4);
    float* accb = (float*)carve((size_t)n_nodes * H_DIM * 4);
    float* x1   = (float*)carve((size_t)n_nodes * H_DIM * 4);
    float* x2   = (float*)carve((size_t)n_nodes * H_DIM * 4);
    float* h3   = (float*)carve((size_t)n_nodes * 4);
    float* out3 = (float*)carve((size_t)n_nodes * 4);
    int*   counts = (int*)carve((size_t)n_graphs * 4);
    int*   starts = (int*)carve((size_t)n_graphs * 4);
    int*   sel    = (int*)carve((size_t)n_graphs * KSEL * 4);
    float* z1     = (float*)carve((size_t)n_graphs * NCH1 * KSEL * 4);
    float* flat   = (float*)carve((size_t)n_graphs * FLAT_DIM * 4);
    float* r1     = (float*)carve((size_t)n_graphs * H_DIM * 4);

    const size_t nodeF   = (size_t)n_nodes * H_DIM;
    const size_t edgeF   = (size_t)n_edges * H_DIM;
    const int    B       = 256;
    auto blks = [](size_t n, int b) { return (unsigned)((n + b - 1) / b); };

    dim3 gemm_grid(512 / 128, (n_nodes + 15) / 16);

    // ---- degree -> dinv ----
    fill_f32<<<blks(n_nodes, B), B, 0, stream>>>(dinv, 1.0f, n_nodes);
    deg_edges<<<blks(n_edges, B), B, 0, stream>>>(dinv, dst, n_edges);
    deg_to_dinv<<<blks(n_nodes, B), B, 0, stream>>>(dinv, n_nodes);

    // ---- layer 1: x1 = elu(gcn(x, W1, b1)) ----
    gemm512_wmma<<<gemm_grid, B, 0, stream>>>(x, W1, hbuf, n_nodes);
    init_acc512<<<blks(nodeF, B), B, 0, stream>>>(accb, hbuf, dinv, b1, n_nodes);
    scatter512<<<blks(edgeF, B), B, 0, stream>>>(hbuf, src, dst, dinv, accb, n_edges);
    elu_kernel<<<blks(nodeF, B), B, 0, stream>>>(x1, accb, nullptr, nodeF);

    // ---- layer 2: x2 = elu(gcn(x1, W2, b2) + x1) ----
    gemm512_wmma<<<gemm_grid, B, 0, stream>>>(x1, W2, hbuf, n_nodes);
    init_acc512<<<blks(nodeF, B), B, 0, stream>>>(accb, hbuf, dinv, b2, n_nodes);
    scatter512<<<blks(edgeF, B), B, 0, stream>>>(hbuf, src, dst, dinv, accb, n_edges);
    elu_kernel<<<blks(nodeF, B), B, 0, stream>>>(x2, accb, x1, nodeF);

    // ---- layer 3 (scalar): out3 = gcn(x2, W3, b3) ----
    matvec_w3<<<blks((size_t)n_nodes * 32, B), B, 0, stream>>>(x2, W3, h3, n_nodes);
    out3_init<<<blks(n_nodes, B), B, 0, stream>>>(out3, h3, dinv, b3, n_nodes);
    scatter_scalar<<<blks(n_edges, B), B, 0, stream>>>(h3, src, dst, dinv, out3, n_edges);

    // ---- sort-pool: per-graph top-30 by out3 (key) ----
    zero_i32<<<blks(n_graphs, B), B, 0, stream>>>(counts, n_graphs);
    hist_batch<<<blks(n_nodes, B), B, 0, stream>>>(batch, counts, n_nodes);
    scan_starts<<<1, 1, 0, stream>>>(counts, starts, n_graphs);
    topk_kernel<<<n_graphs, 32, 0, stream>>>(out3, starts, counts, sel);

    // ---- conv head ----
    conv1_relu<<<blks((size_t)n_graphs * KSEL * NCH1, B), B, 0, stream>>>(
        x1, x2, out3, sel, cw1, cb1, z1, n_graphs);
    conv2_pool_relu<<<blks((size_t)n_graphs * NCH2 * CONV2_LEN, B), B, 0, stream>>>(
        z1, cw2, cb2, flat, n_graphs);
    dense1_relu<<<blks((size_t)n_graphs * H_DIM, B), B, 0, stream>>>(flat, lw1, lb1, r1, n_graphs);
    dense2<<<blks((size_t)n_graphs * NCLS, B), B, 0, stream>>>(r1, lw2, lb2, (float*)d_out, n_graphs);
}